// GNN_80762565034554
// MI455X (gfx1250) — compile-verified
//
#include <hip/hip_runtime.h>

typedef float v2f __attribute__((ext_vector_type(2)));
typedef float v8f __attribute__((ext_vector_type(8)));

#define HID 128
#define GCDIM 192
#define EAD 46
#define NN 50000
#define NE 600000
#define NG 256

__device__ __forceinline__ v8f wmma_f32(v2f a, v2f b, v8f c) {
  // D = A(16x4 f32) * B(4x16 f32) + C(16x16 f32)
  return __builtin_amdgcn_wmma_f32_16x16x4_f32(false, a, false, b, (short)0, c,
                                               false, false);
}

// Wave-private LDS producer->consumer handoff: DS ops are in-order within a
// wave on CDNA5; we only need to stop the compiler from reordering.
__device__ __forceinline__ void lds_fence() {
  __builtin_amdgcn_wave_barrier();
  asm volatile("" ::: "memory");
}

__device__ __forceinline__ void atomicMaxF(float* addr, float val) {
  if (val >= 0.f) atomicMax((int*)addr, __float_as_int(val));
  else            atomicMin((unsigned int*)addr, __float_as_uint(val));
}

// ---------------------------------------------------------------------------
// x0 = qa_emb @ W_qa.T + b_qa ; w_fold = W_edge.T @ att_edge  (1 block, 128 thr)
// ---------------------------------------------------------------------------
__global__ void gnn_precompute(const float* __restrict__ qa,
                               const float* __restrict__ W_qa,
                               const float* __restrict__ b_qa,
                               const float* __restrict__ W_edge,
                               const float* __restrict__ att_edge,
                               float* __restrict__ x0,
                               float* __restrict__ wfold) {
  int t = threadIdx.x;
  float s = b_qa[t];
  for (int j = 0; j < 1024; ++j) s += qa[j] * W_qa[t * 1024 + j];
  x0[t] = s;
  float w = 0.f;
  for (int o = 0; o < HID; ++o) w += att_edge[o] * W_edge[o * HID + t];
  wfold[t] = w;
}

// ---------------------------------------------------------------------------
// Zero/ -inf init of accumulation buffers (runs every launch: deterministic).
// ---------------------------------------------------------------------------
__global__ void gnn_init(float* __restrict__ outn, float* __restrict__ denom,
                         float* __restrict__ m, float* __restrict__ pool,
                         float* __restrict__ counts) {
  for (int gid = blockIdx.x * blockDim.x + threadIdx.x; gid < NN * HID;
       gid += gridDim.x * blockDim.x) {
    outn[gid] = 0.f;
    if (gid < NN) { denom[gid] = 0.f; m[gid] = -__builtin_inff(); }
    if (gid < NG * HID) pool[gid] = 0.f;
    if (gid < NG) counts[gid] = 0.f;
  }
}

// ---------------------------------------------------------------------------
// Node phase: xrelu[16,192] -> h[16,128] via fp32 WMMA ; a_src/a_dst dots.
// Dyn LDS: packed W_gat frags (24576 f) + per-wave xrelu (4 x 16x200 f).
// ---------------------------------------------------------------------------
__global__ void gnn_node(const float* __restrict__ x, const float* __restrict__ x0,
                         const float* __restrict__ ntypes,
                         const float* __restrict__ nscore,
                         const float* __restrict__ W_nt,
                         const float* __restrict__ b_nt,
                         const float* __restrict__ W_gat,
                         const float* __restrict__ att_src,
                         const float* __restrict__ att_dst,
                         float* __restrict__ h, float* __restrict__ a_src,
                         float* __restrict__ a_dst) {
  extern __shared__ float lds[];
  v2f* pk = (v2f*)lds;                               // 12288 x float2
  float* xr = lds + 24576 + (threadIdx.x >> 5) * (16 * 200);
  const int lane = threadIdx.x & 31;
  const int ln = lane & 15, half = lane >> 4;

  // Stage W_gat in B-fragment order: frag(ks,ct) = 32 consecutive float2.
  for (int idx = threadIdx.x; idx < 12288; idx += blockDim.x) {
    int l = idx & 31, ct = (idx >> 5) & 7, ks = idx >> 8;
    int o = ct * 16 + (l & 15);
    int k = ks * 4 + ((l >> 4) << 1);
    v2f w; w.x = W_gat[o * GCDIM + k]; w.y = W_gat[o * GCDIM + k + 1];
    pk[idx] = w;
  }
  __syncthreads();

  int wid = blockIdx.x * (blockDim.x >> 5) + (threadIdx.x >> 5);
  int nw = gridDim.x * (blockDim.x >> 5);
  const int ntiles = NN / 16;  // 3125
  for (int tile = wid; tile < ntiles; tile += nw) {
    int r0 = tile * 16;
    // Build xrelu[16,192] = relu([x|extras]) in LDS (row stride 200).
    for (int idx = lane; idx < 16 * GCDIM; idx += 32) {
      int rr = idx / GCDIM, c = idx - rr * GCDIM;
      int row = r0 + rr;
      float v;
      if (c < HID) {
        v = (row == 0) ? x0[c] : x[row * HID + c];
      } else {
        int c2 = c - HID;
        float s = b_nt[c2];
#pragma unroll
        for (int j = 0; j < 4; ++j) s += ntypes[row * 4 + j] * W_nt[c2 * 5 + j];
        s += nscore[row] * W_nt[c2 * 5 + 4];
        v = s;
      }
      xr[rr * 200 + c] = fmaxf(v, 0.f);
    }
    lds_fence();

    v8f acc[8];
#pragma unroll
    for (int ct = 0; ct < 8; ++ct) acc[ct] = {};
    const v2f* arow = (const v2f*)(xr + ln * 200 + 2 * half);
    for (int ks = 0; ks < 48; ++ks) {
      v2f a = arow[ks * 2];
#pragma unroll
      for (int ct = 0; ct < 8; ++ct)
        acc[ct] = wmma_f32(a, pk[(ks * 8 + ct) * 32 + lane], acc[ct]);
    }

    // Epilogue: store h, accumulate att dots. Row m = v + 8*half, col ct*16+ln.
    float ps[8], pd[8];
#pragma unroll
    for (int v = 0; v < 8; ++v) { ps[v] = 0.f; pd[v] = 0.f; }
#pragma unroll
    for (int ct = 0; ct < 8; ++ct) {
      float as = att_src[ct * 16 + ln];
      float ad = att_dst[ct * 16 + ln];
#pragma unroll
      for (int v = 0; v < 8; ++v) {
        float val = acc[ct][v];
        h[(r0 + v + 8 * half) * HID + ct * 16 + ln] = val;
        ps[v] += val * as;
        pd[v] += val * ad;
      }
    }
#pragma unroll
    for (int v = 0; v < 8; ++v)
#pragma unroll
      for (int msk = 1; msk < 16; msk <<= 1) {
        ps[v] += __shfl_xor(ps[v], msk, 16);
        pd[v] += __shfl_xor(pd[v], msk, 16);
      }
    if (ln == 0) {
#pragma unroll
      for (int v = 0; v < 8; ++v) {
        a_src[r0 + v + 8 * half] = ps[v];
        a_dst[r0 + v + 8 * half] = pd[v];
      }
    }
    lds_fence();
  }
}

// ---------------------------------------------------------------------------
// Edge phase: a_edge = relu(relu(attr@W_e1.T+b1)@W_e2.T+b2) . w_fold
// M=32 per wave (2 A-tiles share every B fragment), software-pipelined
// A/B fragment loads so dscnt waits can overlap with the WMMA pipe.
// Dyn LDS: packed W_e1 (6144 f) + packed W_e2 (16384 f) + b1/b2/wfold +
//          per-wave 32x132 intermediate.  2 waves / block.
// ---------------------------------------------------------------------------
__global__ void gnn_edge(const float* __restrict__ eattr,
                         const float* __restrict__ W_e1,
                         const float* __restrict__ b_e1,
                         const float* __restrict__ W_e2,
                         const float* __restrict__ b_e2,
                         const float* __restrict__ wfold,
                         float* __restrict__ a_edge) {
  extern __shared__ float lds[];
  v2f* pk1 = (v2f*)lds;                      // 3072 x float2 (Kpad 48)
  v2f* pk2 = (v2f*)(lds + 6144);             // 8192 x float2 (K 128)
  float* b1 = lds + 22528;
  float* b2 = lds + 22656;
  float* wf = lds + 22784;
  float* i1 = lds + 22912 + (threadIdx.x >> 5) * (32 * 132);
  const int lane = threadIdx.x & 31;
  const int ln = lane & 15, half = lane >> 4;

  for (int idx = threadIdx.x; idx < 3072; idx += blockDim.x) {
    int l = idx & 31, ct = (idx >> 5) & 7, ks = idx >> 8;
    int o = ct * 16 + (l & 15);
    int k = ks * 4 + ((l >> 4) << 1);
    v2f w;
    w.x = (k < EAD) ? W_e1[o * EAD + k] : 0.f;
    w.y = (k + 1 < EAD) ? W_e1[o * EAD + k + 1] : 0.f;
    pk1[idx] = w;
  }
  for (int idx = threadIdx.x; idx < 8192; idx += blockDim.x) {
    int l = idx & 31, ct = (idx >> 5) & 7, ks = idx >> 8;
    int o = ct * 16 + (l & 15);
    int k = ks * 4 + ((l >> 4) << 1);
    v2f w; w.x = W_e2[o * HID + k]; w.y = W_e2[o * HID + k + 1];
    pk2[idx] = w;
  }
  for (int idx = threadIdx.x; idx < HID; idx += blockDim.x) {
    b1[idx] = b_e1[idx]; b2[idx] = b_e2[idx]; wf[idx] = wfold[idx];
  }
  __syncthreads();

  int wid = blockIdx.x * (blockDim.x >> 5) + (threadIdx.x >> 5);
  int nw = gridDim.x * (blockDim.x >> 5);
  const int ntiles = NE / 32;  // 18750
  for (int tile = wid; tile < ntiles; tile += nw) {
    int e0 = tile * 32;
    const float* arow0 = eattr + (size_t)(e0 + ln) * EAD;
    const float* arow1 = arow0 + (size_t)16 * EAD;

    v8f acc[2][8];
#pragma unroll
    for (int t = 0; t < 2; ++t)
#pragma unroll
      for (int ct = 0; ct < 8; ++ct) acc[t][ct] = {};

    v2f Bc[8], Bn[8], a0, a1, a0n, a1n;

    // ---- Layer 1: [32,48pad] @ W_e1.T, pipelined ----
    {
      int k = 2 * half;
      a0.x = arow0[k]; a0.y = arow0[k + 1];
      a1.x = arow1[k]; a1.y = arow1[k + 1];
    }
#pragma unroll
    for (int ct = 0; ct < 8; ++ct) Bc[ct] = pk1[ct * 32 + lane];
    for (int ks = 0; ks < 12; ++ks) {
      if (ks < 11) {
        int k = (ks + 1) * 4 + 2 * half;
        a0n.x = (k < EAD) ? arow0[k] : 0.f;
        a0n.y = (k + 1 < EAD) ? arow0[k + 1] : 0.f;
        a1n.x = (k < EAD) ? arow1[k] : 0.f;
        a1n.y = (k + 1 < EAD) ? arow1[k + 1] : 0.f;
#pragma unroll
        for (int ct = 0; ct < 8; ++ct)
          Bn[ct] = pk1[((ks + 1) * 8 + ct) * 32 + lane];
      }
#pragma unroll
      for (int ct = 0; ct < 8; ++ct) {
        acc[0][ct] = wmma_f32(a0, Bc[ct], acc[0][ct]);
        acc[1][ct] = wmma_f32(a1, Bc[ct], acc[1][ct]);
      }
      a0 = a0n; a1 = a1n;
#pragma unroll
      for (int ct = 0; ct < 8; ++ct) Bc[ct] = Bn[ct];
    }
#pragma unroll
    for (int t = 0; t < 2; ++t)
#pragma unroll
      for (int ct = 0; ct < 8; ++ct) {
        float bb = b1[ct * 16 + ln];
#pragma unroll
        for (int v = 0; v < 8; ++v)
          i1[(t * 16 + v + 8 * half) * 132 + ct * 16 + ln] =
              fmaxf(acc[t][ct][v] + bb, 0.f);
      }
    lds_fence();

    // ---- Layer 2: [32,128] @ W_e2.T, pipelined ----
#pragma unroll
    for (int t = 0; t < 2; ++t)
#pragma unroll
      for (int ct = 0; ct < 8; ++ct) acc[t][ct] = {};
    const v2f* ar0 = (const v2f*)(i1 + ln * 132 + 2 * half);
    const v2f* ar1 = (const v2f*)(i1 + (16 + ln) * 132 + 2 * half);
    a0 = ar0[0]; a1 = ar1[0];
#pragma unroll
    for (int ct = 0; ct < 8; ++ct) Bc[ct] = pk2[ct * 32 + lane];
    for (int ks = 0; ks < 32; ++ks) {
      if (ks < 31) {
        a0n = ar0[(ks + 1) * 2];
        a1n = ar1[(ks + 1) * 2];
#pragma unroll
        for (int ct = 0; ct < 8; ++ct)
          Bn[ct] = pk2[((ks + 1) * 8 + ct) * 32 + lane];
      }
#pragma unroll
      for (int ct = 0; ct < 8; ++ct) {
        acc[0][ct] = wmma_f32(a0, Bc[ct], acc[0][ct]);
        acc[1][ct] = wmma_f32(a1, Bc[ct], acc[1][ct]);
      }
      a0 = a0n; a1 = a1n;
#pragma unroll
      for (int ct = 0; ct < 8; ++ct) Bc[ct] = Bn[ct];
    }
    lds_fence();
#pragma unroll
    for (int t = 0; t < 2; ++t)
#pragma unroll
      for (int ct = 0; ct < 8; ++ct) {
        float bb = b2[ct * 16 + ln];
#pragma unroll
        for (int v = 0; v < 8; ++v)
          i1[(t * 16 + v + 8 * half) * 132 + ct * 16 + ln] =
              fmaxf(acc[t][ct][v] + bb, 0.f);
      }
    lds_fence();
    {  // a_edge: each of the 32 lanes reduces one row against w_fold
      float s = 0.f;
      for (int c = 0; c < HID; ++c) s += i1[lane * 132 + c] * wf[c];
      a_edge[e0 + lane] = s;
    }
    lds_fence();
  }
}

// ---------------------------------------------------------------------------
// Attention softmax + scatter
// ---------------------------------------------------------------------------
__global__ void gnn_logits_max(const int* __restrict__ ei,
                               const float* __restrict__ a_src,
                               const float* __restrict__ a_dst,
                               const float* __restrict__ a_edge,
                               float* __restrict__ logits,
                               float* __restrict__ m) {
  int e = blockIdx.x * blockDim.x + threadIdx.x;
  if (e >= NE) return;
  int s = ei[e], d = ei[NE + e];
  float z = a_src[s] + a_dst[d] + a_edge[e];
  z = (z > 0.f) ? z : 0.2f * z;  // leaky_relu 0.2
  logits[e] = z;
  atomicMaxF(&m[d], z);
}

__global__ void gnn_exp_sum(const int* __restrict__ ei,
                            const float* __restrict__ logits,
                            const float* __restrict__ m,
                            float* __restrict__ exbuf,
                            float* __restrict__ denom) {
  int e = blockIdx.x * blockDim.x + threadIdx.x;
  if (e >= NE) return;
  int d = ei[NE + e];
  float ex = expf(logits[e] - m[d]);
  exbuf[e] = ex;
  atomicAdd(&denom[d], ex);
}

__global__ void gnn_scatter(const int* __restrict__ ei,
                            const float* __restrict__ exbuf,
                            const float* __restrict__ denom,
                            const float* __restrict__ h,
                            float* __restrict__ outn) {
  int gid = blockIdx.x * blockDim.x + threadIdx.x;
  int e = gid >> 5;
  if (e >= NE) return;
  int lane = gid & 31;
  int s = ei[e], d = ei[NE + e];
  float alpha = exbuf[e] / (denom[d] + 1e-16f);
  float4 hv = ((const float4*)(h + (size_t)s * HID))[lane];
  float* op = outn + (size_t)d * HID + lane * 4;
  atomicAdd(op + 0, alpha * hv.x);
  atomicAdd(op + 1, alpha * hv.y);
  atomicAdd(op + 2, alpha * hv.z);
  atomicAdd(op + 3, alpha * hv.w);
}

__global__ void gnn_pool(const float* __restrict__ outn,
                         const float* __restrict__ b_gat,
                         const int* __restrict__ n2g,
                         float* __restrict__ pool, float* __restrict__ counts) {
  int gid = blockIdx.x * blockDim.x + threadIdx.x;
  if (gid >= NN * HID) return;
  int i = gid >> 7, c = gid & 127;
  float v = outn[gid] + b_gat[c];
  int g = n2g[i];
  atomicAdd(&pool[g * HID + c], v);
  if (c == 0) atomicAdd(&counts[g], 1.f);
}

__global__ void gnn_finalize(const float* __restrict__ outn,
                             const float* __restrict__ b_gat,
                             const float* __restrict__ pool,
                             const float* __restrict__ counts,
                             float* __restrict__ out) {
  int gid = blockIdx.x * blockDim.x + threadIdx.x;
  if (gid < 128) out[gid] = outn[gid] + b_gat[gid];
  if (gid < NG * HID) {
    int g = gid >> 7;
    out[HID + gid] = pool[gid] / fmaxf(counts[g], 1.f);
  }
}

// ---------------------------------------------------------------------------
extern "C" void kernel_launch(void* const* d_in, const int* in_sizes, int n_in,
                              void* d_out, int out_size, void* d_ws,
                              size_t ws_size, hipStream_t stream) {
  const float* qa_emb   = (const float*)d_in[0];
  const float* x        = (const float*)d_in[1];
  const float* ntypes   = (const float*)d_in[3];
  const float* nscore   = (const float*)d_in[4];
  const int*   ei       = (const int*)d_in[5];
  const float* eattr    = (const float*)d_in[7];
  const int*   n2g      = (const int*)d_in[8];
  const float* W_qa     = (const float*)d_in[9];
  const float* b_qa     = (const float*)d_in[10];
  const float* W_nt     = (const float*)d_in[11];
  const float* b_nt     = (const float*)d_in[12];
  const float* W_e1     = (const float*)d_in[13];
  const float* b_e1     = (const float*)d_in[14];
  const float* W_e2     = (const float*)d_in[15];
  const float* b_e2     = (const float*)d_in[16];
  const float* W_gat    = (const float*)d_in[17];
  const float* W_edge   = (const float*)d_in[18];
  const float* att_src  = (const float*)d_in[19];
  const float* att_dst  = (const float*)d_in[20];
  const float* att_edge = (const float*)d_in[21];
  const float* b_gat    = (const float*)d_in[22];
  float* out = (float*)d_out;

  // Workspace layout (floats)
  float* ws = (float*)d_ws;
  size_t o_h    = 0;
  size_t o_out  = o_h + (size_t)NN * HID;
  size_t o_asrc = o_out + (size_t)NN * HID;
  size_t o_adst = o_asrc + NN;
  size_t o_m    = o_adst + NN;
  size_t o_den  = o_m + NN;
  size_t o_ae   = o_den + NN;       // a_edge, reused as exp buffer
  size_t o_log  = o_ae + NE;
  size_t o_x0   = o_log + NE;
  size_t o_wf   = o_x0 + 128;
  size_t o_pool = o_wf + 128;
  size_t o_cnt  = o_pool + (size_t)NG * HID;
  float* w_h = ws + o_h;   float* w_out = ws + o_out;
  float* w_as = ws + o_asrc; float* w_ad = ws + o_adst;
  float* w_m = ws + o_m;   float* w_den = ws + o_den;
  float* w_ae = ws + o_ae; float* w_log = ws + o_log;
  float* w_x0 = ws + o_x0; float* w_wf = ws + o_wf;
  float* w_pool = ws + o_pool; float* w_cnt = ws + o_cnt;

  gnn_precompute<<<1, 128, 0, stream>>>(qa_emb, W_qa, b_qa, W_edge, att_edge,
                                        w_x0, w_wf);
  gnn_init<<<4096, 256, 0, stream>>>(w_out, w_den, w_m, w_pool, w_cnt);

  // Node GEMM: dyn LDS = packed W_gat (96KB) + 4 waves x 16x200 xrelu (51.2KB)
  size_t node_lds = (24576 + 4 * 16 * 200) * sizeof(float);
  gnn_node<<<256, 128, node_lds, stream>>>(x, w_x0, ntypes, nscore, W_nt, b_nt,
                                           W_gat, att_src, att_dst, w_h, w_as,
                                           w_ad);

  // Edge MLP: dyn LDS = packed W_e1 (24KB) + W_e2 (64KB) + vecs + 2x32x132
  size_t edge_lds = (22912 + 2 * 32 * 132) * sizeof(float);
  gnn_edge<<<1024, 64, edge_lds, stream>>>(eattr, W_e1, b_e1, W_e2, b_e2, w_wf,
                                           w_ae);

  int eb = (NE + 255) / 256;
  gnn_logits_max<<<eb, 256, 0, stream>>>(ei, w_as, w_ad, w_ae, w_log, w_m);
  gnn_exp_sum<<<eb, 256, 0, stream>>>(ei, w_log, w_m, w_ae, w_den);
  gnn_scatter<<<(NE * 32 + 255) / 256, 256, 0, stream>>>(ei, w_ae, w_den, w_h,
                                                         w_out);
  gnn_pool<<<(NN * HID + 255) / 256, 256, 0, stream>>>(w_out, b_gat, n2g,
                                                       w_pool, w_cnt);
  gnn_finalize<<<(NG * HID + 255) / 256, 256, 0, stream>>>(w_out, b_gat, w_pool,
                                                           w_cnt, out);
  (void)in_sizes; (void)n_in; (void)out_size; (void)ws_size;
}